// Pointnet_SA_MSG_24378234372449
// MI455X (gfx1250) — compile-verified
//
#include <hip/hip_runtime.h>
#include <hip/hip_bf16.h>

typedef __attribute__((ext_vector_type(2))) float v2f;
typedef __attribute__((ext_vector_type(8))) float v8f;

#define NPTS   4096
#define NCENT  1024
#define NBATCH 8
#define FEATC  320

// ---------------------------------------------------------------------------
// zero-fill output
// ---------------------------------------------------------------------------
__global__ void zero_kernel(float* __restrict__ p, int n) {
    int i = blockIdx.x * 256 + threadIdx.x;
    if (i < n) p[i] = 0.0f;
}

// ---------------------------------------------------------------------------
// Farthest point sampling: one workgroup per batch, dists in LDS.
// Matches jnp.argmax first-max tie-breaking (prefer smaller index).
// ---------------------------------------------------------------------------
__global__ __launch_bounds__(256) void fps_kernel(const float* __restrict__ xyz,
                                                  float* __restrict__ new_xyz) {
    __shared__ float dists[NPTS];
    __shared__ float rv[256];
    __shared__ int   ri[256];
    __shared__ int   sfar;
    const int b   = blockIdx.x;
    const int tid = threadIdx.x;
    const float* px = xyz + (size_t)b * NPTS * 3;

    for (int i = tid; i < NPTS; i += 256) dists[i] = 1e10f;
    __syncthreads();

    int farthest = 0;
    for (int it = 0; it < NCENT; ++it) {
        if (tid == 0) {
            new_xyz[((size_t)b * NCENT + it) * 3 + 0] = px[farthest * 3 + 0];
            new_xyz[((size_t)b * NCENT + it) * 3 + 1] = px[farthest * 3 + 1];
            new_xyz[((size_t)b * NCENT + it) * 3 + 2] = px[farthest * 3 + 2];
        }
        const float cx = px[farthest * 3 + 0];
        const float cy = px[farthest * 3 + 1];
        const float cz = px[farthest * 3 + 2];
        float bv = -1.0f; int bi = 0;
        for (int i = tid; i < NPTS; i += 256) {
            const float dx = px[i * 3 + 0] - cx;
            const float dy = px[i * 3 + 1] - cy;
            const float dz = px[i * 3 + 2] - cz;
            const float d  = dx * dx + dy * dy + dz * dz;
            const float nd = fminf(dists[i], d);
            dists[i] = nd;
            if (nd > bv) { bv = nd; bi = i; }   // strided i increasing -> first-max kept
        }
        rv[tid] = bv; ri[tid] = bi;
        __syncthreads();
        for (int s = 128; s > 0; s >>= 1) {
            if (tid < s) {
                const float v2 = rv[tid + s]; const int i2 = ri[tid + s];
                if (v2 > rv[tid] || (v2 == rv[tid] && i2 < ri[tid])) { rv[tid] = v2; ri[tid] = i2; }
            }
            __syncthreads();
        }
        if (tid == 0) sfar = ri[0];
        __syncthreads();
        farthest = sfar;
    }
}

// ---------------------------------------------------------------------------
// Ball query: one wave per centroid; ordered compaction of first-ns in-radius
// indices for all three radii in one scan; pad with first hit.
// ---------------------------------------------------------------------------
__global__ __launch_bounds__(32) void ballquery_kernel(const float* __restrict__ xyz,
                                                       const float* __restrict__ new_xyz,
                                                       int* __restrict__ gi0,
                                                       int* __restrict__ gi1,
                                                       int* __restrict__ gi2) {
    const int bp   = blockIdx.x;          // b*1024 + p
    const int b    = bp >> 10;
    const int lane = threadIdx.x;
    const float cx = new_xyz[bp * 3 + 0];
    const float cy = new_xyz[bp * 3 + 1];
    const float cz = new_xyz[bp * 3 + 2];
    const float* px = xyz + (size_t)b * NPTS * 3;

    const float R0 = (float)(0.1 * 0.1);
    const float R1 = (float)(0.2 * 0.2);
    const float R2 = (float)(0.4 * 0.4);

    int* o0 = gi0 + (size_t)bp * 16;
    int* o1 = gi1 + (size_t)bp * 32;
    int* o2 = gi2 + (size_t)bp * 128;

    int c0 = 0, c1 = 0, c2 = 0;
    int f0 = 0, f1 = 0, f2 = 0;
    const unsigned long long lt = (1ull << lane) - 1ull;

    for (int base = 0; base < NPTS; base += 32) {
        const int i = base + lane;
        const float dx = px[i * 3 + 0] - cx;
        const float dy = px[i * 3 + 1] - cy;
        const float dz = px[i * 3 + 2] - cz;
        const float d2 = dx * dx + dy * dy + dz * dz;

        { // scale 0
            const unsigned long long m = __ballot(d2 < R0);
            if (m != 0ull && c0 < 16) {
                if (c0 == 0) f0 = base + __builtin_ctzll(m);
                const int slot = c0 + (int)__popcll(m & lt);
                if ((d2 < R0) && slot < 16) o0[slot] = i;
            }
            if (c0 < 16) c0 += (int)__popcll(m);
        }
        { // scale 1
            const unsigned long long m = __ballot(d2 < R1);
            if (m != 0ull && c1 < 32) {
                if (c1 == 0) f1 = base + __builtin_ctzll(m);
                const int slot = c1 + (int)__popcll(m & lt);
                if ((d2 < R1) && slot < 32) o1[slot] = i;
            }
            if (c1 < 32) c1 += (int)__popcll(m);
        }
        { // scale 2
            const unsigned long long m = __ballot(d2 < R2);
            if (m != 0ull && c2 < 128) {
                if (c2 == 0) f2 = base + __builtin_ctzll(m);
                const int slot = c2 + (int)__popcll(m & lt);
                if ((d2 < R2) && slot < 128) o2[slot] = i;
            }
            if (c2 < 128) c2 += (int)__popcll(m);
        }
    }
    for (int j = min(c0, 16)  + lane; j < 16;  j += 32) o0[j] = f0;
    for (int j = min(c1, 32)  + lane; j < 32;  j += 32) o1[j] = f1;
    for (int j = min(c2, 128) + lane; j < 128; j += 32) o2[j] = f2;
}

// ---------------------------------------------------------------------------
// Per-wave fp32 GEMM tile via V_WMMA_F32_16X16X4_F32:
//   Y(16 x COUT) = relu( X(16 x KD) @ W(KD x COUT) + bias ), X padded to KP.
// A layout (16x4 f32): lane<16 -> K={k0,k0+1}; lane>=16 -> K={k0+2,k0+3}.
// B layout (4x16 f32): same K split, N = lane%16.
// C/D layout: VGPR j -> row j + 8*(lane/16), col = lane%16.
// Padding rows (KD <= k < KP) are handled branch-free: clamped-row load +
// select-to-zero, so EXEC stays all-ones with no divergent-load exec juggling.
// ---------------------------------------------------------------------------
template<int KP, int KD, int COUT>
__device__ __forceinline__ void wave_gemm_relu(const float* __restrict__ inb,
                                               const float* __restrict__ wg,
                                               const float* __restrict__ bg,
                                               float* __restrict__ outb,
                                               int lane) {
    const int half = lane >> 4;
    const int l16  = lane & 15;
    for (int n0 = 0; n0 < COUT; n0 += 16) {
        v8f acc = {0.f, 0.f, 0.f, 0.f, 0.f, 0.f, 0.f, 0.f};
        for (int k0 = 0; k0 < KP; k0 += 4) {
            const int ka = k0 + half * 2;
            // A fragment: two contiguous, 8B-aligned floats -> single ds_load_b64
            const v2f af = *(const v2f*)(inb + l16 * KP + ka);
            v2f bf;
            if constexpr (KD == KP) {
                bf.x = wg[(size_t)(ka    ) * COUT + n0 + l16];
                bf.y = wg[(size_t)(ka + 1) * COUT + n0 + l16];
            } else {
                const int r0 = (ka     < KD) ? ka     : (KD - 1);
                const int r1 = (ka + 1 < KD) ? ka + 1 : (KD - 1);
                const float t0 = wg[(size_t)r0 * COUT + n0 + l16];
                const float t1 = wg[(size_t)r1 * COUT + n0 + l16];
                bf.x = (ka     < KD) ? t0 : 0.0f;
                bf.y = (ka + 1 < KD) ? t1 : 0.0f;
            }
            acc = __builtin_amdgcn_wmma_f32_16x16x4_f32(
                false, af, false, bf, (short)0, acc, false, false);
        }
        const float bias = bg[n0 + l16];
#pragma unroll
        for (int j = 0; j < 8; ++j) {
            float v = acc[j] + bias;
            outb[(j + half * 8) * COUT + (n0 + l16)] = v > 0.0f ? v : 0.0f;
        }
    }
}

// ---------------------------------------------------------------------------
// Fused group -> 3-layer MLP -> maxpool. One wave per 16-sample tile.
// ---------------------------------------------------------------------------
template<int NS, int C1, int C2, int C3, int FOFF>
__global__ __launch_bounds__(32) void mlp_kernel(const float* __restrict__ xyz,
                                                 const float* __restrict__ points,
                                                 const float* __restrict__ new_xyz,
                                                 const int*   __restrict__ idx,
                                                 const float* __restrict__ w0, const float* __restrict__ b0,
                                                 const float* __restrict__ w1, const float* __restrict__ b1,
                                                 const float* __restrict__ w2, const float* __restrict__ b2,
                                                 float* __restrict__ outf) {
    constexpr int KP0 = 68;                 // 67 input channels padded to mult-of-4
    __shared__ float bufA[16 * 128];
    __shared__ float bufB[16 * 128];
    const int lane  = threadIdx.x;
    constexpr int TILES = NS / 16;
    const int gid = blockIdx.x;
    const int t   = gid % TILES;
    const int bp  = gid / TILES;            // b*1024 + p
    const int b   = bp >> 10;

    const float cx = new_xyz[bp * 3 + 0];
    const float cy = new_xyz[bp * 3 + 1];
    const float cz = new_xyz[bp * 3 + 2];
    const int* ip = idx + (size_t)bp * NS + t * 16;

    // gather 16 rows of [points(64) | xyz-rel(3) | pad(1)] into LDS
    for (int r = 0; r < 16; ++r) {
        const int gi = ip[r];
        const float* src = points + ((size_t)b * NPTS + gi) * 64;
        bufA[r * KP0 + lane]      = src[lane];
        bufA[r * KP0 + lane + 32] = src[lane + 32];
        if (lane == 0) {
            const float* pxp = xyz + ((size_t)b * NPTS + gi) * 3;
            bufA[r * KP0 + 64] = pxp[0] - cx;
            bufA[r * KP0 + 65] = pxp[1] - cy;
            bufA[r * KP0 + 66] = pxp[2] - cz;
            bufA[r * KP0 + 67] = 0.0f;
        }
    }
    __syncthreads();
    wave_gemm_relu<KP0, 67, C1>(bufA, w0, b0, bufB, lane);
    __syncthreads();
    wave_gemm_relu<C1, C1, C2>(bufB, w1, b1, bufA, lane);
    __syncthreads();
    wave_gemm_relu<C2, C2, C3>(bufA, w2, b2, bufB, lane);
    __syncthreads();

    // max over 16 rows; combine tiles via bitwise atomicMax (values >= 0)
    float* dst = outf + (size_t)bp * FEATC + FOFF;
    for (int ch = lane; ch < C3; ch += 32) {
        float m = bufB[ch];
#pragma unroll
        for (int r = 1; r < 16; ++r) m = fmaxf(m, bufB[r * C3 + ch]);
        atomicMax((int*)&dst[ch], __float_as_int(m));
    }
}

// ---------------------------------------------------------------------------
extern "C" void kernel_launch(void* const* d_in, const int* in_sizes, int n_in,
                              void* d_out, int out_size, void* d_ws, size_t ws_size,
                              hipStream_t stream) {
    const float* xyz = (const float*)d_in[0];
    const float* pts = (const float*)d_in[1];
    const float* w00 = (const float*)d_in[2];
    const float* b00 = (const float*)d_in[3];
    const float* w01 = (const float*)d_in[4];
    const float* b01 = (const float*)d_in[5];
    const float* w02 = (const float*)d_in[6];
    const float* b02 = (const float*)d_in[7];
    const float* w10 = (const float*)d_in[8];
    const float* b10 = (const float*)d_in[9];
    const float* w11 = (const float*)d_in[10];
    const float* b11 = (const float*)d_in[11];
    const float* w12 = (const float*)d_in[12];
    const float* b12 = (const float*)d_in[13];
    const float* w20 = (const float*)d_in[14];
    const float* b20 = (const float*)d_in[15];
    const float* w21 = (const float*)d_in[16];
    const float* b21 = (const float*)d_in[17];
    const float* w22 = (const float*)d_in[18];
    const float* b22 = (const float*)d_in[19];

    float* out     = (float*)d_out;
    float* new_xyz = out;                               // B*1024*3
    float* feat    = out + (size_t)NBATCH * NCENT * 3;  // B*1024*320

    int* i0 = (int*)d_ws;
    int* i1 = i0 + (size_t)NBATCH * NCENT * 16;
    int* i2 = i1 + (size_t)NBATCH * NCENT * 32;

    const int total = NBATCH * NCENT * 3 + NBATCH * NCENT * FEATC;
    zero_kernel<<<(total + 255) / 256, 256, 0, stream>>>(out, total);

    fps_kernel<<<NBATCH, 256, 0, stream>>>(xyz, new_xyz);

    ballquery_kernel<<<NBATCH * NCENT, 32, 0, stream>>>(xyz, new_xyz, i0, i1, i2);

    mlp_kernel<16, 32, 32, 64, 0>
        <<<NBATCH * NCENT * 1, 32, 0, stream>>>(xyz, pts, new_xyz, i0,
                                                w00, b00, w01, b01, w02, b02, feat);
    mlp_kernel<32, 64, 64, 128, 64>
        <<<NBATCH * NCENT * 2, 32, 0, stream>>>(xyz, pts, new_xyz, i1,
                                                w10, b10, w11, b11, w12, b12, feat);
    mlp_kernel<128, 64, 96, 128, 192>
        <<<NBATCH * NCENT * 8, 32, 0, stream>>>(xyz, pts, new_xyz, i2,
                                                w20, b20, w21, b21, w22, b22, feat);
}